// GraphModule_50165218017605
// MI455X (gfx1250) — compile-verified
//
#include <hip/hip_runtime.h>
#include <cstdint>

typedef __attribute__((ext_vector_type(2))) float v2f;
typedef __attribute__((ext_vector_type(8))) float v8f;

#define HD 128          // hidden / feature dim (F = H = OUT = 128)
#define BN_EPS 1e-5f
#define NEG_SLOPE 0.1f

// ---------------------------------------------------------------------------
// WMMA GEMM: C[M x 128] = A[M x 128] @ B[128 x 128] (+ optional bias)
// Block = 256 threads (8 waves).  grid.x = row group (128 rows),
// grid.y = column tile (16 cols).  Wave w owns rows [128*bx + 16w, +16):
// A loads are non-redundant across waves.  The 128x16 W column slab (8KB)
// is staged in LDS once per block (conflict-free frag reads).
// Each wave: 32x v_wmma_f32_16x16x4_f32 over K=128.
// A-frag layout (32-bit A 16x4): lanes 0-15 hold K=k0,k0+1; lanes 16-31 hold
// K=k0+2,k0+3 for rows M=0..15.  C layout: VGPR v, lanes 0-15 -> M=v,
// lanes 16-31 -> M=v+8, N=lane%16.
// ---------------------------------------------------------------------------
__global__ __launch_bounds__(256) void gemm128_wmma(
    const float* __restrict__ A, const float* __restrict__ B,
    const float* __restrict__ bias, float* __restrict__ C, int M)
{
    __shared__ float Bs[HD * 16];     // W[:, col0:col0+16], Bs[k*16 + c]
    const int tid  = threadIdx.x;
    const int col0 = blockIdx.y * 16;

    // stage the 128x16 column slab: 2048 floats / 256 threads = 8 each
#pragma unroll
    for (int it = 0; it < (HD * 16) / 256; ++it) {
        int j = tid + it * 256;       // j = k*16 + c
        int k = j >> 4, c = j & 15;
        Bs[j] = B[k * HD + col0 + c];
    }
    __syncthreads();

    const int wave = tid >> 5;
    const int lane = tid & 31;
    const int half = lane >> 4;       // 0: K pair (k0,k0+1), 1: (k0+2,k0+3)
    const int l16  = lane & 15;
    const int row0 = blockIdx.x * 128 + wave * 16;

    int arow = row0 + l16;
    if (arow >= M) arow = M - 1;      // branchless clamp (EXEC stays all-1s)
    const float* ap = A + (size_t)arow * HD + 2 * half;

    v8f acc = {};
#pragma unroll
    for (int k0 = 0; k0 < HD; k0 += 4) {
        v2f a = *(const v2f*)(ap + k0);                 // 8B-aligned global b64
        v2f b;
        b.x = Bs[(k0 + 2 * half) * 16 + l16];
        b.y = Bs[(k0 + 2 * half + 1) * 16 + l16];
        acc = __builtin_amdgcn_wmma_f32_16x16x4_f32(
            /*neg_a=*/false, a, /*neg_b=*/false, b,
            /*c_mod=*/(short)0, acc, /*reuse_a=*/false, /*reuse_b=*/false);
    }

    const float bv = bias ? bias[col0 + l16] : 0.0f;
#pragma unroll
    for (int v = 0; v < 8; ++v) {
        const int r = row0 + v + 8 * half;
        if (r < M) C[(size_t)r * HD + col0 + l16] = acc[v] + bv;
    }
}

// ---------------------------------------------------------------------------
// Small helper kernels
// ---------------------------------------------------------------------------
__global__ void k_fill(float* __restrict__ p, float val, int n)
{
    int i = blockIdx.x * blockDim.x + threadIdx.x;
    if (i < n) p[i] = val;
}

__global__ void k_deg_count(const long long* __restrict__ col,
                            float* __restrict__ deg, int E)
{
    int e = blockIdx.x * blockDim.x + threadIdx.x;
    if (e < E) atomicAdd(&deg[(int)col[e]], 1.0f);
}

__global__ void k_rsqrt_inplace(float* __restrict__ p, int n)
{
    int i = blockIdx.x * blockDim.x + threadIdx.x;
    if (i < n) p[i] = rsqrtf(p[i]);
}

// agg[n,f] = 2*dinv[n]^2 * xw[n,f] + bias[f]   (improved self-loop + bias)
__global__ void k_agg_init(const float* __restrict__ xw,
                           const float* __restrict__ dinv,
                           const float* __restrict__ bias,
                           float* __restrict__ agg, int NM)
{
    int i = blockIdx.x * blockDim.x + threadIdx.x;
    if (i >= NM) return;
    int n = i >> 7, f = i & (HD - 1);
    float di = dinv[n];
    agg[i] = 2.0f * di * di * xw[i] + bias[f];
}

// one wave per EPW (4) consecutive edges; lane handles 4 consecutive features (float4 gather,
// 4x global_atomic_add_f32 scatter)
#define EPW 4
__global__ __launch_bounds__(256) void k_edge_scatter(
    const long long* __restrict__ row, const long long* __restrict__ col,
    const float* __restrict__ dinv, const float* __restrict__ xw,
    float* __restrict__ agg, int E)
{
    const int lane = threadIdx.x & 31;
    const int wid  = blockIdx.x * (blockDim.x >> 5) + (threadIdx.x >> 5);
    const int e0   = wid * EPW;
#pragma unroll
    for (int i = 0; i < EPW; ++i) {
        int e = e0 + i;               // wave-uniform
        if (e >= E) return;
        int r = (int)row[e];
        int c = (int)col[e];
        float w = dinv[r] * dinv[c];
        float4 xv = ((const float4*)(xw + (size_t)r * HD))[lane];
        float* dst = agg + (size_t)c * HD + lane * 4;
        atomicAdd(dst + 0, w * xv.x);
        atomicAdd(dst + 1, w * xv.y);
        atomicAdd(dst + 2, w * xv.z);
        atomicAdd(dst + 3, w * xv.w);
    }
}

// per-feature sum / sumsq partials -> stats[0:128]=sum, stats[128:256]=sumsq
__global__ __launch_bounds__(128) void k_bn_stats(const float* __restrict__ h,
                                                  float* __restrict__ stats, int N)
{
    int f = threadIdx.x;
    float s = 0.f, sq = 0.f;
    for (int n = blockIdx.x; n < N; n += gridDim.x) {
        float v = h[(size_t)n * HD + f];
        s += v; sq += v * v;
    }
    atomicAdd(&stats[f], s);
    atomicAdd(&stats[HD + f], sq);
}

// h = leaky_relu((h - mu) * rsqrt(var + eps) * w + b)
__global__ void k_bn_lrelu(float* __restrict__ h, const float* __restrict__ stats,
                           const float* __restrict__ w, const float* __restrict__ b,
                           int N, int NM)
{
    int i = blockIdx.x * blockDim.x + threadIdx.x;
    if (i >= NM) return;
    int f = i & (HD - 1);
    float invN = 1.0f / (float)N;
    float mu   = stats[f] * invN;
    float var  = stats[HD + f] * invN - mu * mu;
    float sc   = w[f] * rsqrtf(var + BN_EPS);
    float sh   = b[f] - mu * sc;
    float v    = h[i] * sc + sh;
    h[i] = v > 0.f ? v : NEG_SLOPE * v;
}

// float atomic max via sign-split int/uint trick (monotone bit orderings)
__device__ __forceinline__ void atomicMaxF(float* addr, float v)
{
    if (v >= 0.f) atomicMax((int*)addr, __float_as_int(v));
    else          atomicMin((unsigned int*)addr, __float_as_uint(v));
}

__global__ void k_pool_max(const float* __restrict__ h,
                           const long long* __restrict__ batch,
                           float* __restrict__ pooled, int NM)
{
    int i = blockIdx.x * blockDim.x + threadIdx.x;
    if (i >= NM) return;
    int n = i >> 7, f = i & (HD - 1);
    int g = (int)batch[n];
    atomicMaxF(&pooled[(size_t)g * HD + f], h[i]);
}

__global__ void k_pool_fixup(float* __restrict__ pooled, int n)
{
    int i = blockIdx.x * blockDim.x + threadIdx.x;
    if (i < n) {
        float v = pooled[i];
        pooled[i] = (v == -__builtin_inff()) ? 0.f : v;
    }
}

// ---------------------------------------------------------------------------
extern "C" void kernel_launch(void* const* d_in, const int* in_sizes, int n_in,
                              void* d_out, int out_size, void* d_ws, size_t ws_size,
                              hipStream_t stream)
{
    const float*     x     = (const float*)d_in[0];
    const long long* ei    = (const long long*)d_in[1];
    const long long* batch = (const long long*)d_in[2];
    const float*     W1    = (const float*)d_in[3];
    const float*     b1    = (const float*)d_in[4];
    const float*     bn1w  = (const float*)d_in[5];
    const float*     bn1b  = (const float*)d_in[6];
    const float*     W2    = (const float*)d_in[7];
    const float*     b2    = (const float*)d_in[8];
    const float*     bn2w  = (const float*)d_in[9];
    const float*     bn2b  = (const float*)d_in[10];
    const float*     Wf    = (const float*)d_in[11];
    const float*     bf    = (const float*)d_in[12];

    const int N  = in_sizes[0] / HD;
    const int E  = in_sizes[1] / 2;
    const int G  = out_size / HD;
    const int NM = N * HD;
    const long long* row = ei;
    const long long* col = ei + E;

    // workspace carve-out (floats)
    float* ws     = (float*)d_ws;
    float* xw     = ws;                       // N*HD
    float* agg    = xw + (size_t)N * HD;      // N*HD
    float* dinv   = agg + (size_t)N * HD;     // N (holds deg, then rsqrt(deg))
    float* stats  = dinv + N;                 // 2*HD
    float* pooled = stats + 2 * HD;           // G*HD

    const int T = 256;
    auto cdiv = [](int a, int b) { return (a + b - 1) / b; };
    const int edgesPerBlock = (256 / 32) * EPW;   // 32

    dim3 gemmGridN(cdiv(N, 128), HD / 16);    // rows x col-tiles
    dim3 gemmGridG(cdiv(G, 128), HD / 16);

    // ---- degrees: deg = 2 + indegree; dinv = rsqrt(deg) ----
    k_fill<<<cdiv(N, T), T, 0, stream>>>(dinv, 2.0f, N);
    k_deg_count<<<cdiv(E, T), T, 0, stream>>>(col, dinv, E);
    k_rsqrt_inplace<<<cdiv(N, T), T, 0, stream>>>(dinv, N);

    // ---- layer 1 ----
    gemm128_wmma<<<gemmGridN, 256, 0, stream>>>(x, W1, nullptr, xw, N);
    k_agg_init<<<cdiv(NM, T), T, 0, stream>>>(xw, dinv, b1, agg, NM);
    k_edge_scatter<<<cdiv(E, edgesPerBlock), 256, 0, stream>>>(row, col, dinv, xw, agg, E);
    k_fill<<<1, 256, 0, stream>>>(stats, 0.f, 2 * HD);
    k_bn_stats<<<512, 128, 0, stream>>>(agg, stats, N);
    k_bn_lrelu<<<cdiv(NM, T), T, 0, stream>>>(agg, stats, bn1w, bn1b, N, NM);

    // ---- layer 2 (reuse buffers: xw = h1@W2, agg overwritten) ----
    gemm128_wmma<<<gemmGridN, 256, 0, stream>>>(agg, W2, nullptr, xw, N);
    k_agg_init<<<cdiv(NM, T), T, 0, stream>>>(xw, dinv, b2, agg, NM);
    k_edge_scatter<<<cdiv(E, edgesPerBlock), 256, 0, stream>>>(row, col, dinv, xw, agg, E);
    k_fill<<<1, 256, 0, stream>>>(stats, 0.f, 2 * HD);
    k_bn_stats<<<512, 128, 0, stream>>>(agg, stats, N);
    k_bn_lrelu<<<cdiv(NM, T), T, 0, stream>>>(agg, stats, bn2w, bn2b, N, NM);

    // ---- global max pool + final linear ----
    k_fill<<<cdiv(G * HD, T), T, 0, stream>>>(pooled, -__builtin_inff(), G * HD);
    k_pool_max<<<cdiv(NM, T), T, 0, stream>>>(agg, batch, pooled, NM);
    k_pool_fixup<<<cdiv(G * HD, T), T, 0, stream>>>(pooled, G * HD);
    gemm128_wmma<<<gemmGridG, 256, 0, stream>>>(pooled, Wf, bf, (float*)d_out, G);
}